// GCN_41798621725026
// MI455X (gfx1250) — compile-verified
//
#include <hip/hip_runtime.h>
#include <math.h>

// GCN forward: x[N,128] -> GCNConv(16) -> ELU -> GCNConv(32) -> ELU
//              -> scatter-mean over graphs -> Linear(32,1)
// N=100000, E=3200000, G=1000. Bandwidth/scatter bound; WMMA f32 for GEMMs.

typedef float v2f __attribute__((ext_vector_type(2)));
typedef float v8f __attribute__((ext_vector_type(8)));

#define N_NODES  100000
#define N_EDGES  3200000
#define F_INDIM  128
#define H1DIM    16
#define H2DIM    32
#define N_GRAPHS 1000

// ---------------------------------------------------------------- utilities
__global__ void fill_f32(float* __restrict__ p, float v, int n) {
    int i = blockIdx.x * blockDim.x + threadIdx.x;
    int stride = gridDim.x * blockDim.x;
    for (; i < n; i += stride) p[i] = v;
}

// deg[dst] += 1 per directed edge (deg pre-initialized to 1.0 == self-loop)
__global__ void deg_accum(const int* __restrict__ dst, float* __restrict__ deg, int e) {
    int i = blockIdx.x * blockDim.x + threadIdx.x;
    if (i < e) atomicAdd(&deg[dst[i]], 1.0f);
}

__global__ void dinv_kernel(const float* __restrict__ deg, float* __restrict__ dinv, int n) {
    int i = blockIdx.x * blockDim.x + threadIdx.x;
    if (i < n) {
        float d = deg[i];
        dinv[i] = (d > 0.0f) ? rsqrtf(d) : 0.0f;
    }
}

// ----------------------------------------------------- GEMM1: h1 = x @ w1
// x[N,128] (row-major) * w1[128,16] -> h1[N,16]. One wave = 16 rows.
// V_WMMA_F32_16X16X4_F32, 32 k-steps of 4.
__global__ __launch_bounds__(256) void gemm1_wmma(const float* __restrict__ x,
                                                  const float* __restrict__ w1,
                                                  float* __restrict__ h1, int n) {
    int wave = blockIdx.x * (blockDim.x >> 5) + (threadIdx.x >> 5);
    int lane = threadIdx.x & 31;
    int rowBase = wave * 16;
    if (rowBase >= n) return;            // wave-uniform: EXEC stays all-1s

    int half = lane >> 4;                // 0: lanes 0-15, 1: lanes 16-31
    int l    = lane & 15;

    const float* xrow = x + (size_t)(rowBase + l) * F_INDIM;
    v8f c = {};
    #pragma unroll 4
    for (int k = 0; k < F_INDIM; k += 4) {
        // A 16x4 f32: VGPR0 = K(2*half), VGPR1 = K(2*half+1), row = l
        v2f a;
        a.x = xrow[k + 2 * half + 0];
        a.y = xrow[k + 2 * half + 1];
        // B 4x16 f32: VGPR0 = row K(2*half), VGPR1 = row K(2*half+1), col = l
        v2f b;
        b.x = w1[(k + 2 * half + 0) * H1DIM + l];
        b.y = w1[(k + 2 * half + 1) * H1DIM + l];
        c = __builtin_amdgcn_wmma_f32_16x16x4_f32(false, a, false, b,
                                                  (short)0, c, false, false);
    }
    // C/D 16x16 f32: VGPR r -> row (r + 8*half), col = l
    #pragma unroll
    for (int r = 0; r < 8; ++r)
        h1[(size_t)(rowBase + r + 8 * half) * H1DIM + l] = c[r];
}

// ------------------------------------------ GEMM2: h2 = h1e @ w2 (16 -> 32)
__global__ __launch_bounds__(256) void gemm2_wmma(const float* __restrict__ h1e,
                                                  const float* __restrict__ w2,
                                                  float* __restrict__ h2, int n) {
    int wave = blockIdx.x * (blockDim.x >> 5) + (threadIdx.x >> 5);
    int lane = threadIdx.x & 31;
    int rowBase = wave * 16;
    if (rowBase >= n) return;

    int half = lane >> 4;
    int l    = lane & 15;

    const float* arow = h1e + (size_t)(rowBase + l) * H1DIM;
    v8f c0 = {}, c1 = {};
    #pragma unroll
    for (int k = 0; k < H1DIM; k += 4) {
        v2f a;
        a.x = arow[k + 2 * half + 0];
        a.y = arow[k + 2 * half + 1];
        v2f b0, b1;   // two 16-wide column tiles of w2[16,32]
        b0.x = w2[(k + 2 * half + 0) * H2DIM + l];
        b0.y = w2[(k + 2 * half + 1) * H2DIM + l];
        b1.x = w2[(k + 2 * half + 0) * H2DIM + 16 + l];
        b1.y = w2[(k + 2 * half + 1) * H2DIM + 16 + l];
        c0 = __builtin_amdgcn_wmma_f32_16x16x4_f32(false, a, false, b0,
                                                   (short)0, c0, false, false);
        c1 = __builtin_amdgcn_wmma_f32_16x16x4_f32(false, a, false, b1,
                                                   (short)0, c1, false, false);
    }
    #pragma unroll
    for (int r = 0; r < 8; ++r) {
        size_t row = (size_t)(rowBase + r + 8 * half);
        h2[row * H2DIM + l]      = c0[r];
        h2[row * H2DIM + 16 + l] = c1[r];
    }
}

// ---------------------------------------- edge scatter: agg[dst] += norm*h[src]
__global__ void scatter16(const int* __restrict__ src, const int* __restrict__ dst,
                          const float* __restrict__ dinv, const float* __restrict__ h,
                          float* __restrict__ agg, int e) {
    int i = blockIdx.x * blockDim.x + threadIdx.x;
    if (i >= e) return;
    int s = src[i], d = dst[i];
    float nrm = dinv[s] * dinv[d];
    const float4* hv = (const float4*)(h + (size_t)s * H1DIM);   // 64B rows
    float* ad = agg + (size_t)d * H1DIM;
    #pragma unroll
    for (int q = 0; q < 4; ++q) {
        float4 v = hv[q];
        atomicAdd(ad + 4 * q + 0, v.x * nrm);
        atomicAdd(ad + 4 * q + 1, v.y * nrm);
        atomicAdd(ad + 4 * q + 2, v.z * nrm);
        atomicAdd(ad + 4 * q + 3, v.w * nrm);
    }
}

__global__ void scatter32(const int* __restrict__ src, const int* __restrict__ dst,
                          const float* __restrict__ dinv, const float* __restrict__ h,
                          float* __restrict__ agg, int e) {
    int i = blockIdx.x * blockDim.x + threadIdx.x;
    if (i >= e) return;
    int s = src[i], d = dst[i];
    float nrm = dinv[s] * dinv[d];
    const float4* hv = (const float4*)(h + (size_t)s * H2DIM);   // 128B rows
    float* ad = agg + (size_t)d * H2DIM;
    #pragma unroll
    for (int q = 0; q < 8; ++q) {
        float4 v = hv[q];
        atomicAdd(ad + 4 * q + 0, v.x * nrm);
        atomicAdd(ad + 4 * q + 1, v.y * nrm);
        atomicAdd(ad + 4 * q + 2, v.z * nrm);
        atomicAdd(ad + 4 * q + 3, v.w * nrm);
    }
}

// ---------------- epilogue: add self-loop (dinv^2 * h), bias, ELU (in place)
__global__ void post_elu(float* __restrict__ agg, const float* __restrict__ h,
                         const float* __restrict__ dinv, const float* __restrict__ bias,
                         int n, int fdim) {
    int idx = blockIdx.x * blockDim.x + threadIdx.x;
    int total = n * fdim;
    if (idx >= total) return;
    int i = idx / fdim;
    int f = idx - i * fdim;
    float di = dinv[i];
    float v = agg[idx] + di * di * h[idx] + bias[f];
    agg[idx] = (v > 0.0f) ? v : (expf(v) - 1.0f);
}

// ----------------------------- scatter-mean pooling accumulation per node
__global__ void pool_accum(const float* __restrict__ h2e, const int* __restrict__ batch,
                           float* __restrict__ pool, float* __restrict__ cnt, int n) {
    int i = blockIdx.x * blockDim.x + threadIdx.x;
    if (i >= n) return;
    int g = batch[i];
    atomicAdd(&cnt[g], 1.0f);
    const float* hr = h2e + (size_t)i * H2DIM;
    float* pg = pool + (size_t)g * H2DIM;
    #pragma unroll
    for (int f = 0; f < H2DIM; ++f) atomicAdd(pg + f, hr[f]);
}

// ----------------------------------------- final: out[g] = mean @ w3 + b3
__global__ void final_linear(const float* __restrict__ pool, const float* __restrict__ cnt,
                             const float* __restrict__ w3, const float* __restrict__ b3,
                             float* __restrict__ out, int g) {
    int i = blockIdx.x * blockDim.x + threadIdx.x;
    if (i >= g) return;
    float c = fmaxf(cnt[i], 1.0f);
    float inv = 1.0f / c;
    const float* pg = pool + (size_t)i * H2DIM;
    float s = 0.0f;
    #pragma unroll
    for (int f = 0; f < H2DIM; ++f) s += pg[f] * w3[f];
    out[i] = s * inv + b3[0];
}

// ---------------------------------------------------------------- launcher
extern "C" void kernel_launch(void* const* d_in, const int* in_sizes, int n_in,
                              void* d_out, int out_size, void* d_ws, size_t ws_size,
                              hipStream_t stream) {
    const float* x     = (const float*)d_in[0];
    const int*   ei    = (const int*)d_in[1];
    const int*   batch = (const int*)d_in[2];
    const float* w1    = (const float*)d_in[3];
    const float* b1    = (const float*)d_in[4];
    const float* w2    = (const float*)d_in[5];
    const float* b2    = (const float*)d_in[6];
    const float* w3    = (const float*)d_in[7];
    const float* b3    = (const float*)d_in[8];
    float* out = (float*)d_out;

    const int n = N_NODES, e = N_EDGES, g = N_GRAPHS;
    const int* src = ei;         // edge_index[0,:]
    const int* dst = ei + e;     // edge_index[1,:]

    // workspace carve-out (floats); ~39.3 MB total
    float* ws   = (float*)d_ws;
    float* deg  = ws;  ws += n;
    float* dinv = ws;  ws += n;
    float* h1   = ws;  ws += (size_t)n * H1DIM;
    float* agg1 = ws;  ws += (size_t)n * H1DIM;   // becomes h1e after post_elu
    float* h2   = ws;  ws += (size_t)n * H2DIM;
    float* agg2 = ws;  ws += (size_t)n * H2DIM;   // becomes h2e after post_elu
    float* pool = ws;  ws += (size_t)g * H2DIM;
    float* cnt  = ws;  ws += g;

    const int TB = 256;
    auto blocks = [](int work, int tb) { return (work + tb - 1) / tb; };

    // init (harness poisons ws; must zero every call)
    fill_f32<<<blocks(n, TB), TB, 0, stream>>>(deg, 1.0f, n);          // self-loop
    fill_f32<<<blocks(n * H1DIM, TB), TB, 0, stream>>>(agg1, 0.0f, n * H1DIM);
    fill_f32<<<blocks(n * H2DIM, TB), TB, 0, stream>>>(agg2, 0.0f, n * H2DIM);
    fill_f32<<<blocks(g * H2DIM, TB), TB, 0, stream>>>(pool, 0.0f, g * H2DIM);
    fill_f32<<<blocks(g, TB), TB, 0, stream>>>(cnt, 0.0f, g);

    // degrees + normalization
    deg_accum<<<blocks(e, TB), TB, 0, stream>>>(dst, deg, e);
    dinv_kernel<<<blocks(n, TB), TB, 0, stream>>>(deg, dinv, n);

    // layer 1: GEMM (WMMA f32) -> edge scatter -> self-loop+bias+ELU
    int waves1 = (n + 15) / 16;                       // 6250 waves
    int gblk   = (waves1 + (TB / 32) - 1) / (TB / 32);
    gemm1_wmma<<<gblk, TB, 0, stream>>>(x, w1, h1, n);
    scatter16<<<blocks(e, TB), TB, 0, stream>>>(src, dst, dinv, h1, agg1, e);
    post_elu<<<blocks(n * H1DIM, TB), TB, 0, stream>>>(agg1, h1, dinv, b1, n, H1DIM);

    // layer 2
    gemm2_wmma<<<gblk, TB, 0, stream>>>(agg1, w2, h2, n);
    scatter32<<<blocks(e, TB), TB, 0, stream>>>(src, dst, dinv, h2, agg2, e);
    post_elu<<<blocks(n * H2DIM, TB), TB, 0, stream>>>(agg2, h2, dinv, b2, n, H2DIM);

    // pooling + readout
    pool_accum<<<blocks(n, TB), TB, 0, stream>>>(agg2, batch, pool, cnt, n);
    final_linear<<<blocks(g, TB), TB, 0, stream>>>(pool, cnt, w3, b3, out, g);
}